// XentLoss_77455440216461
// MI455X (gfx1250) — compile-verified
//
#include <hip/hip_runtime.h>
#include <cmath>

// Label-smoothed cross entropy: rows = B*T = 4096, V = 32000.
// Memory-bound streaming reduction; WMMA f32 16x16x4 used for the
// 256-partial block reduction (CDNA5 matrix path).

typedef __attribute__((ext_vector_type(2))) float v2f;
typedef __attribute__((ext_vector_type(4))) float v4f;
typedef __attribute__((ext_vector_type(8))) float v8f;

#define THREADS 256

__global__ __launch_bounds__(THREADS)
void XentLoss_row_kernel(const float* __restrict__ lp,
                         const int* __restrict__ trg,
                         float* __restrict__ row_out,
                         int V,
                         float conf, float smv, float log_conf, float log_sm) {
    const int row = blockIdx.x;
    const size_t row_base = (size_t)row * (size_t)V;
    const v4f* __restrict__ rp = reinterpret_cast<const v4f*>(lp + row_base);
    const int nvec = V >> 2;   // V divisible by 4 (32000 -> 8000)

    // ---- Phase 1: coalesced non-temporal streaming row sum (HBM-bound) ----
    float acc = 0.0f;
    for (int i = threadIdx.x; i < nvec; i += THREADS) {
        // prefetch next tile of the stream (speculative, OOB is dropped)
        __builtin_prefetch(reinterpret_cast<const char*>(rp + i + THREADS), 0, 0);
        v4f v = __builtin_nontemporal_load(rp + i);   // global_load_b128, TH=NT
        acc += (v[0] + v[1]) + (v[2] + v[3]);
    }

    __shared__ float partials[THREADS];
    __shared__ float s_row_sum;
    partials[threadIdx.x] = acc;
    __syncthreads();

    // ---- Phase 2: reduce 256 partials with V_WMMA_F32_16X16X4_F32 ----
    // Treat partials as a 16x16 f32 A-matrix; B = all-ones 4x16.
    // D[m][n] = sum_k A[m][k] + C[m][n]  (4 chunks of K=4 accumulate into C).
    if (threadIdx.x < 32) {             // wave 0, EXEC all ones (uniform branch)
        const int lane = threadIdx.x;
        const int m    = lane & 15;
        const int koff = (lane >> 4) << 1;   // 0 for lanes 0-15, 2 for 16-31
        v8f c = {};
        v2f b; b[0] = 1.0f; b[1] = 1.0f;     // ones fragment (layout-invariant)
#if defined(__HIP_DEVICE_COMPILE__)
#pragma unroll
        for (int ch = 0; ch < 4; ++ch) {
            v2f a;
            a[0] = partials[m * 16 + ch * 4 + koff];
            a[1] = partials[m * 16 + ch * 4 + koff + 1];
            c = __builtin_amdgcn_wmma_f32_16x16x4_f32(
                    /*neg_a=*/false, a, /*neg_b=*/false, b,
                    /*c_mod=*/(short)0, c, /*reuse_a=*/false, /*reuse_b=*/false);
        }
#else
        (void)b;
        for (int k = 0; k < 16; ++k) c[0] += partials[m * 16 + k]; // host parse only
#endif
        // VGPR r holds D row r (lanes<16) / r+8 (lanes>=16); every column equals
        // that row's sum. Sum the 8 accumulator regs, then fold halves.
        float s = ((c[0] + c[1]) + (c[2] + c[3])) + ((c[4] + c[5]) + (c[6] + c[7]));
        s += __shfl_xor(s, 16, 32);          // rows 0-7 + rows 8-15
        if (lane == 0) s_row_sum = s;
    }
    __syncthreads();

    // ---- Phase 3: per-row loss ----
    if (threadIdx.x == 0) {
        const float row_sum = s_row_sum;
        const int   t       = trg[row];
        const float lp_t    = lp[row_base + (size_t)t];
        const float lp_pad  = lp[row_base];          // PAD_INDEX == 0
        float r = conf * (log_conf - lp_t)
                + smv * (float)(V - 2) * log_sm
                - smv * (row_sum - lp_t - lp_pad);
        row_out[row] = (t != 0) ? r : 0.0f;          // mask folded in
    }
}

__global__ __launch_bounds__(THREADS)
void XentLoss_final_kernel(const float* __restrict__ row_vals,
                           float* __restrict__ out, int rows) {
    __shared__ float smem[THREADS];
    float acc = 0.0f;
    for (int i = threadIdx.x; i < rows; i += THREADS)
        acc += row_vals[i];
    smem[threadIdx.x] = acc;
    __syncthreads();
    for (int off = THREADS >> 1; off > 0; off >>= 1) {
        if (threadIdx.x < off) smem[threadIdx.x] += smem[threadIdx.x + off];
        __syncthreads();
    }
    if (threadIdx.x == 0) out[0] = smem[0];
}

extern "C" void kernel_launch(void* const* d_in, const int* in_sizes, int n_in,
                              void* d_out, int out_size, void* d_ws, size_t ws_size,
                              hipStream_t stream) {
    const float* lp  = (const float*)d_in[0];
    const int*   trg = (const int*)d_in[1];
    const int rows = in_sizes[1];                 // B*T = 4096
    const int V    = in_sizes[0] / in_sizes[1];   // 32000

    const double conf_d = 1.0 - 0.1;              // SMOOTHING = 0.1
    const double sm_d   = 0.1 / (double)(V - 2);
    const float conf     = (float)conf_d;
    const float smv      = (float)sm_d;
    const float log_conf = (float)log(conf_d);
    const float log_sm   = (float)log(sm_d);

    float* row_vals = (float*)d_ws;               // rows * 4 bytes (16 KB)

    XentLoss_row_kernel<<<rows, THREADS, 0, stream>>>(
        lp, trg, row_vals, V, conf, smv, log_conf, log_sm);
    XentLoss_final_kernel<<<1, THREADS, 0, stream>>>(row_vals, (float*)d_out, rows);
}